// YoloLoss_34711925687028
// MI455X (gfx1250) — compile-verified
//
#include <hip/hip_runtime.h>

// ---------------------------------------------------------------------------
// YOLO loss, MI455X (gfx1250, wave32).
// Bandwidth-bound elementwise + reduction; WMMA f32 16x16x4 (A = ones) used as
// an exact in-matrix-pipe block reducer. Deterministic 2-stage reduction.
// ---------------------------------------------------------------------------

#define NCELLS (4096 * 14 * 14)   // 802816 = 3136 * 256 exactly
#define NBLK   3136
#define SGRID  14.0f
#define INVN   (1.0f / 4096.0f)

typedef __attribute__((ext_vector_type(2))) float v2f;
typedef __attribute__((ext_vector_type(8))) float v8f;

struct Box { float x1, y1, x2, y2; };

__device__ __forceinline__ Box xywh2xyxy(float x, float y, float w, float h) {
    Box b;
    float xs = x / SGRID, ys = y / SGRID;
    b.x1 = xs - 0.5f * w; b.y1 = ys - 0.5f * h;
    b.x2 = xs + 0.5f * w; b.y2 = ys + 0.5f * h;
    return b;
}

__device__ __forceinline__ float iou_fn(const Box& a, const Box& b) {
    float ltx = fmaxf(a.x1, b.x1), lty = fmaxf(a.y1, b.y1);
    float rbx = fminf(a.x2, b.x2), rby = fminf(a.y2, b.y2);
    float w = fmaxf(rbx - ltx, 0.0f), h = fmaxf(rby - lty, 0.0f);
    float inter = w * h;
    float aa = (a.x2 - a.x1) * (a.y2 - a.y1);
    float ab = (b.x2 - b.x1) * (b.y2 - b.y1);
    return inter / (aa + ab - inter);
}

// Reduce 4 per-thread partials across a 256-thread block using
// v_wmma_f32_16x16x4_f32 with A = ones (exact f32 column sums).
// Component c of column n is c = n & 3; 16 accumulating WMMAs cover all
// 4*256 LDS slots. Returns component sum for tid < 4 (component == tid).
__device__ __forceinline__ float block_reduce4_wmma(const float p[4],
                                                    float* lds /*4*256*/,
                                                    float* red /*16*/) {
    const int tid = threadIdx.x;
#pragma unroll
    for (int c = 0; c < 4; ++c) lds[c * 256 + tid] = p[c];
    __syncthreads();

    if (tid < 32) {                 // whole wave 0, EXEC all-ones for WMMA
        const int L    = tid;
        const int Khi  = (L >> 4) << 1;       // K rows {Khi, Khi+1} per ISA B layout
        const int N    = L & 15;
        const int base = (N & 3) * 256 + (N >> 2);
        v2f ones; ones[0] = 1.0f; ones[1] = 1.0f;
        v8f acc = {};
#pragma unroll
        for (int it = 0; it < 16; ++it) {
            v2f b;
            b[0] = lds[base + 4 * (Khi + 0) + 16 * it];
            b[1] = lds[base + 4 * (Khi + 1) + 16 * it];
            // D[m][n] = sum_k 1 * B[k][n] + C[m][n]  (exact f32 adds)
            acc = __builtin_amdgcn_wmma_f32_16x16x4_f32(
                false, ones, false, b, (short)0, acc, false, false);
        }
        // Row M=0 of C/D lives in VGPR0, lanes 0..15 (N = lane).
        if (L < 16) red[L] = acc[0];
    }
    __syncthreads();
    float s = 0.0f;
    if (tid < 4) s = red[tid] + red[tid + 4] + red[tid + 8] + red[tid + 12];
    return s;
}

__global__ __launch_bounds__(256) void yolo_partials(
    const float* __restrict__ pred, const float* __restrict__ tbox,
    const float* __restrict__ tcls, const int* __restrict__ hasobj,
    float* __restrict__ part) {
    __shared__ float lds[4 * 256];
    __shared__ float red[16];

    const int tid  = threadIdx.x;
    const int cell = blockIdx.x * 256 + tid;
    float p[4] = {0.0f, 0.0f, 0.0f, 0.0f};   // class, no_obj, reg, cont

    if (cell < NCELLS) {
        // pred: 30 f32 per cell, 8B-aligned per cell -> 15 x float2
        const float2* pr2 = reinterpret_cast<const float2*>(pred) + (size_t)cell * 15;
        float pv[30];
#pragma unroll
        for (int i = 0; i < 15; ++i) { float2 t = pr2[i]; pv[2*i] = t.x; pv[2*i+1] = t.y; }
        // target_cls: 20 f32, 16B-aligned -> 5 x float4
        const float4* tc4 = reinterpret_cast<const float4*>(tcls) + (size_t)cell * 5;
        float tcv[20];
#pragma unroll
        for (int i = 0; i < 5; ++i) {
            float4 t = tc4[i];
            tcv[4*i] = t.x; tcv[4*i+1] = t.y; tcv[4*i+2] = t.z; tcv[4*i+3] = t.w;
        }
        const float4 tb = reinterpret_cast<const float4*>(tbox)[cell];
        const float  m  = hasobj[cell] ? 1.0f : 0.0f;

        // class loss: mask * sum((pred[10:30]-tcls)^2)
        float cl = 0.0f;
#pragma unroll
        for (int j = 0; j < 20; ++j) { float d = pv[10 + j] - tcv[j]; cl += d * d; }
        p[0] = m * cl;

        // no-object loss: (1-mask) * (pred4^2 + pred9^2)
        p[1] = (1.0f - m) * (pv[4] * pv[4] + pv[9] * pv[9]);

        // boxes (note: reference applies xywh2xyxy to the target TWICE -> t2)
        Box b0 = xywh2xyxy(pv[0], pv[1], pv[2], pv[3]);
        Box b1 = xywh2xyxy(pv[5], pv[6], pv[7], pv[8]);
        Box t1 = xywh2xyxy(tb.x, tb.y, tb.z, tb.w);
        Box t2 = xywh2xyxy(t1.x1, t1.y1, t1.x2, t1.y2);

        float i0 = iou_fn(b0, t1);
        float i1 = iou_fn(b1, t2);
        bool  pick1 = i1 > i0;
        float best  = fmaxf(i0, i1);

        float bx = pick1 ? b1.x1 : b0.x1;
        float by = pick1 ? b1.y1 : b0.y1;
        float dx = bx - t2.x1, dy = by - t2.y1;
        p[2] = m * (dx * dx + dy * dy);

        float cp = pick1 ? pv[9] : pv[4];
        float dc = cp - best;
        p[3] = m * dc * dc;
    }

    float s = block_reduce4_wmma(p, lds, red);
    if (tid < 4) part[blockIdx.x * 4 + tid] = s;
}

__global__ __launch_bounds__(256) void yolo_final(const float* __restrict__ part,
                                                  float* __restrict__ out) {
    __shared__ float lds[4 * 256];
    __shared__ float red[16];
    __shared__ float comp[4];

    const int tid = threadIdx.x;
    float p[4] = {0.0f, 0.0f, 0.0f, 0.0f};
    for (int i = tid; i < NBLK; i += 256) {
        float4 v = reinterpret_cast<const float4*>(part)[i];
        p[0] += v.x; p[1] += v.y; p[2] += v.z; p[3] += v.w;
    }

    float s = block_reduce4_wmma(p, lds, red);
    if (tid < 4) comp[tid] = s * INVN;
    __syncthreads();

    if (tid == 0) {
        const float cls   = comp[0];
        const float noobj = comp[1];
        const float reg   = comp[2];
        const float cont  = comp[3];
        out[0] = cls + noobj + reg + cont;   // total (same add order as reference)
        out[1] = reg;
        out[2] = cont;
        out[3] = noobj;
        out[4] = cls;
    }
}

extern "C" void kernel_launch(void* const* d_in, const int* in_sizes, int n_in,
                              void* d_out, int out_size, void* d_ws, size_t ws_size,
                              hipStream_t stream) {
    const float* pred   = (const float*)d_in[0];
    const float* tbox   = (const float*)d_in[1];
    const float* tcls   = (const float*)d_in[2];
    const int*   hasobj = (const int*)d_in[3];
    float* part = (float*)d_ws;      // NBLK*4 floats of scratch, rewritten each call
    float* out  = (float*)d_out;

    yolo_partials<<<NBLK, 256, 0, stream>>>(pred, tbox, tcls, hasobj, part);
    yolo_final<<<1, 256, 0, stream>>>(part, out);
    (void)in_sizes; (void)n_in; (void)out_size; (void)ws_size;
}